// MultiHeadEdgeAttention_ParallelBetweenBondtypes_81200651698822
// MI455X (gfx1250) — compile-verified
//
#include <hip/hip_runtime.h>

typedef float v2f __attribute__((ext_vector_type(2)));
typedef float v8f __attribute__((ext_vector_type(8)));

#define NH   8
#define DH   64
#define NT   5
#define NN   256
#define LDSP 68   // padded LDS row stride in floats (16B-aligned, conflict-free)

// One workgroup: (b, h, 64-row i-tile) x all 256 j. 8 waves: 4 row-waves x 2 col-waves.
__global__ __launch_bounds__(256)
void edge_attn_wmma_kernel(const float* __restrict__ src,
                           const float* __restrict__ dst,
                           const float* __restrict__ a,
                           const int*   __restrict__ edges,
                           float* __restrict__ out,
                           int B) {
  extern __shared__ float smem[];
  float* s_lds = smem;               // 64 x LDSP : S tile  [i][d]
  float* a_lds = smem + 64 * LDSP;   // 64 x LDSP : A_t^T   [e][d]
  float* w_lds = smem + 2 * 64 * LDSP; // 64 x LDSP : W_t  [i][e]
  float* d_lds = smem + 3 * 64 * LDSP; // 256 x LDSP : D   [j][e]

  const int tid  = threadIdx.x;
  const int lane = tid & 31;
  const int wid  = tid >> 5;
  const int wr   = wid & 3;    // wave row: 16 i-rows each
  const int wc   = wid >> 2;   // wave col: 128 j-cols each
  const int lo   = lane & 15;
  const int hi   = lane >> 4;

  const int it = blockIdx.x;   // i tile (0..3)
  const int h  = blockIdx.y;   // head
  const int b  = blockIdx.z;   // batch
  const int i0 = it * 64;

  // ---- stage S (64x64) and D (256x64) tiles into LDS (float4, coalesced) ----
  const float* srcp = src + ((size_t)b * NN) * (NH * DH) + h * DH;
  const float* dstp = dst + ((size_t)b * NN) * (NH * DH) + h * DH;
#pragma unroll
  for (int c = 0; c < 4; ++c) {
    int flat = c * 1024 + tid * 4;
    int row = flat >> 6, col = flat & 63;
    float4 v = *(const float4*)(srcp + (size_t)(i0 + row) * (NH * DH) + col);
    *(float4*)(s_lds + row * LDSP + col) = v;
  }
#pragma unroll
  for (int c = 0; c < 16; ++c) {
    int flat = c * 1024 + tid * 4;
    int row = flat >> 6, col = flat & 63;
    float4 v = *(const float4*)(dstp + (size_t)row * (NH * DH) + col);
    *(float4*)(d_lds + row * LDSP + col) = v;
  }

  // ---- per-thread edge codes + output accumulators (C-fragment ownership) ----
  int   ec[8][8];
  float ov[8][8];
  const int* ep = edges + ((size_t)b * NN + i0) * NN;
#pragma unroll
  for (int r = 0; r < 8; ++r) {
#pragma unroll
    for (int f = 0; f < 8; ++f) {
      int ii = wr * 16 + r + 8 * hi;
      int jj = wc * 128 + f * 16 + lo;
      ec[r][f] = ep[(size_t)ii * NN + jj];
      ov[r][f] = 0.0f;
    }
  }

  const float* ap = a + ((size_t)(h * DH)) * (DH * NT); // a[(h*64+dd)][e][t]

  for (int t = 0; t < NT; ++t) {
    // stage A_t transposed: a_lds[e][dd] = a[(h*64+dd)*320 + e*5 + t]
#pragma unroll
    for (int c = 0; c < 16; ++c) {
      int q = c * 256 + tid;
      int dd = q >> 6, e = q & 63;
      a_lds[e * LDSP + dd] = ap[(size_t)dd * (DH * NT) + e * NT + t];
    }
    __syncthreads();

    // ---- phase 1: W_t = S @ A_t (64x64x64); each wave: 16 x 32 ----
    {
      v8f acc0 = {0.f,0.f,0.f,0.f,0.f,0.f,0.f,0.f};
      v8f acc1 = {0.f,0.f,0.f,0.f,0.f,0.f,0.f,0.f};
      const int am  = wr * 16 + lo;   // A frag: M = lane%16
      const int bn0 = wc * 32 + lo;   // B frag: N = lane%16
#pragma unroll
      for (int k = 0; k < 64; k += 4) {
        int kk = k + 2 * hi;          // K pair per half-wave
        v2f av = *(const v2f*)(s_lds + am * LDSP + kk);
        v2f b0 = *(const v2f*)(a_lds + bn0 * LDSP + kk);
        v2f b1 = *(const v2f*)(a_lds + (bn0 + 16) * LDSP + kk);
        acc0 = __builtin_amdgcn_wmma_f32_16x16x4_f32(false, av, false, b0,
                                                     (short)0, acc0, false, false);
        acc1 = __builtin_amdgcn_wmma_f32_16x16x4_f32(false, av, false, b1,
                                                     (short)0, acc1, false, false);
      }
#pragma unroll
      for (int r = 0; r < 8; ++r) {
        w_lds[(wr * 16 + r + 8 * hi) * LDSP + wc * 32 + lo]      = acc0[r];
        w_lds[(wr * 16 + r + 8 * hi) * LDSP + wc * 32 + 16 + lo] = acc1[r];
      }
    }
    __syncthreads();

    // ---- phase 2: Score_t = W_t @ D^T (64x256x64); each wave: 16 x 128 ----
    {
      v8f acc[8];
#pragma unroll
      for (int f = 0; f < 8; ++f) acc[f] = (v8f){0.f,0.f,0.f,0.f,0.f,0.f,0.f,0.f};
      const int am = wr * 16 + lo;
#pragma unroll
      for (int k = 0; k < 64; k += 4) {
        int kk = k + 2 * hi;
        v2f av = *(const v2f*)(w_lds + am * LDSP + kk);
#pragma unroll
        for (int f = 0; f < 8; ++f) {
          v2f bv = *(const v2f*)(d_lds + (wc * 128 + f * 16 + lo) * LDSP + kk);
          acc[f] = __builtin_amdgcn_wmma_f32_16x16x4_f32(false, av, false, bv,
                                                         (short)0, acc[f], false, false);
        }
      }
      // select the bond-type slice per edge
#pragma unroll
      for (int f = 0; f < 8; ++f)
#pragma unroll
        for (int r = 0; r < 8; ++r)
          if (ec[r][f] == t) ov[r][f] = acc[f][r];
    }
    __syncthreads();
  }

  // ---- mask + leaky relu + store ----
  float* op = out + (((size_t)b * NH + h) * NN + i0) * NN;
#pragma unroll
  for (int r = 0; r < 8; ++r) {
#pragma unroll
    for (int f = 0; f < 8; ++f) {
      int ii = wr * 16 + r + 8 * hi;
      int jj = wc * 128 + f * 16 + lo;
      float v = (ec[r][f] == -1) ? -1.0e10f : ov[r][f];
      op[(size_t)ii * NN + jj] = (v > 0.0f) ? v : 0.2f * v;
    }
  }
}

extern "C" void kernel_launch(void* const* d_in, const int* in_sizes, int n_in,
                              void* d_out, int out_size, void* d_ws, size_t ws_size,
                              hipStream_t stream) {
  const float* src   = (const float*)d_in[0];
  const float* dst   = (const float*)d_in[1];
  const float* a     = (const float*)d_in[2];
  const int*   edges = (const int*)d_in[3];
  float* out = (float*)d_out;

  int B = in_sizes[3] / (NN * NN);  // edges: (B, N, N)

  dim3 grid(NN / 64, NH, B);
  dim3 block(256);
  size_t shmem = (size_t)(3 * 64 + 256) * LDSP * sizeof(float); // ~122 KB
  edge_attn_wmma_kernel<<<grid, block, shmem, stream>>>(src, dst, a, edges, out, B);
}